// SS2D_75883482186202
// MI455X (gfx1250) — compile-verified
//
#include <hip/hip_runtime.h>
#include <math.h>

// ---------------------------------------------------------------------------
// SS2D (VMamba selective scan 2D) for MI455X / gfx1250, fp32 end-to-end.
// Dense projections use V_WMMA_F32_16X16X4_F32; the serial selective scan is
// a wave32 kernel with state resident in VGPRs and v_readlane-broadcast B/C/dt.
// ---------------------------------------------------------------------------

typedef __attribute__((ext_vector_type(2))) float v2f;
typedef __attribute__((ext_vector_type(8))) float v8f;

#define BB   8
#define HH   64
#define WW   64
#define CC   96
#define DIN  192
#define NSTATE 16
#define RR   6
#define KDIR 4
#define LL   (HH*WW)          // 4096
#define MROWS (BB*LL)         // 32768
#define CDBL 38               // R + 2N

static __device__ __forceinline__ v8f wmma_f32_k4(v2f a, v2f b, v8f c) {
  // 8-arg form: (neg_a, A, neg_b, B, c_mod, C, reuse_a, reuse_b)
  return __builtin_amdgcn_wmma_f32_16x16x4_f32(false, a, false, b, (short)0, c,
                                               false, false);
}

// Constant-lane broadcast via v_readlane_b32 (SGPR result, no LDS pipe).
static __device__ __forceinline__ float bcast(float v, int srcLane) {
  return __builtin_bit_cast(
      float, __builtin_amdgcn_readlane(__builtin_bit_cast(int, v), srcLane));
}

// -------------------- Kernel 1: in_proj GEMM ------------------------------
// xz[M,384] = x[M,96] @ in_proj_w[384,96]^T     (M = B*L)
__global__ __launch_bounds__(256) void k_inproj(const float* __restrict__ x,
                                                const float* __restrict__ w,
                                                float* __restrict__ xz) {
  const int lane = threadIdx.x & 31;
  const int wid  = blockIdx.x * 8 + (threadIdx.x >> 5);
  const int TN   = (2*DIN)/16;                 // 24 col tiles
  const int tn = wid % TN, tm = wid / TN;
  const int row0 = tm*16, col0 = tn*16;
  const int mrow = lane & 15, half = lane >> 4;
  const float* arow = x + (size_t)(row0 + mrow) * CC;
  const float* brow = w + (size_t)(col0 + mrow) * CC;
  v8f acc = {};
#pragma unroll
  for (int k0 = 0; k0 < CC; k0 += 4) {
    v2f a = *(const v2f*)(arow + k0 + 2*half);
    v2f b = *(const v2f*)(brow + k0 + 2*half);
    acc = wmma_f32_k4(a, b, acc);
  }
  float* out = xz + (size_t)row0 * (2*DIN) + col0 + mrow;
#pragma unroll
  for (int i = 0; i < 8; ++i)
    out[(size_t)(8*half + i) * (2*DIN)] = acc[i];
}

// -------------------- Kernel 2: depthwise conv 3x3 + SiLU^2 ---------------
// reads xi = xz[:, 0:192]; writes xc in [B, L, Din] (channel-last).
__global__ __launch_bounds__(256) void k_conv(const float* __restrict__ xz,
                                              const float* __restrict__ cw,
                                              const float* __restrict__ cb,
                                              float* __restrict__ xc) {
  const int idx = blockIdx.x * 256 + threadIdx.x;     // (b*L + l)*Din + d
  if (idx >= MROWS * DIN) return;
  const int d  = idx % DIN;
  const int bl = idx / DIN;
  const int l  = bl % LL, b = bl / LL;
  const int hh = l / WW, ww = l % WW;
  float acc = cb[d];
#pragma unroll
  for (int dy = -1; dy <= 1; ++dy) {
    const int hy = hh + dy;
    if (hy < 0 || hy >= HH) continue;
#pragma unroll
    for (int dx = -1; dx <= 1; ++dx) {
      const int wx = ww + dx;
      if (wx < 0 || wx >= WW) continue;
      const float v = xz[((size_t)b*LL + hy*WW + wx) * (2*DIN) + d];
      acc = fmaf(v, cw[d*9 + (dy+1)*3 + (dx+1)], acc);
    }
  }
  const float s1 = acc / (1.f + __expf(-acc));
  xc[idx] = s1 / (1.f + __expf(-s1));
}

// direction-dependent spatial position for sequence index l
static __device__ __forceinline__ int dir_pos(int k, int l) {
  if (k == 0) return l;
  if (k == 1) return (l % HH) * WW + (l / HH);
  if (k == 2) return LL - 1 - l;
  const int s = LL - 1 - l;
  return (s % HH) * WW + (s / HH);
}

// -------------------- Kernel 3: per-direction 38-ch projection (GEMM) -----
// xdbl[b,k,l,c] = sum_d x_proj_weight[k,c,d] * xs[b,k,d,l]
__global__ __launch_bounds__(256) void k_xdbl(const float* __restrict__ xc,
                                              const float* __restrict__ wx,
                                              float* __restrict__ xdbl) {
  const int lane = threadIdx.x & 31;
  int wid = blockIdx.x * 8 + (threadIdx.x >> 5);
  const int tn = wid % (LL/16); wid /= (LL/16);      // 256 col tiles (l)
  const int tm = wid % 3;       wid /= 3;            // 3 row tiles (c, 48 pad)
  const int k  = wid % KDIR;
  const int b  = wid / KDIR;
  const int mrow = lane & 15, half = lane >> 4;
  const int row0 = tm*16, col0 = tn*16;
  const int m = row0 + mrow;                         // c row for A loads
  const int l = col0 + mrow;                         // l column for B loads
  const int pos = dir_pos(k, l);
  const bool aok = (m < CDBL);
  const float* arow = wx + ((size_t)k*CDBL + (aok ? m : 0)) * DIN;
  const float* brow = xc + ((size_t)b*LL + pos) * DIN;
  v8f acc = {};
#pragma unroll 8
  for (int k0 = 0; k0 < DIN; k0 += 4) {
    v2f a = {0.f, 0.f};
    if (aok) a = *(const v2f*)(arow + k0 + 2*half);
    v2f bv = *(const v2f*)(brow + k0 + 2*half);
    acc = wmma_f32_k4(a, bv, acc);
  }
  float* out = xdbl + (((size_t)(b*KDIR + k) * LL) + l) * CDBL;
#pragma unroll
  for (int i = 0; i < 8; ++i) {
    const int c = row0 + 8*half + i;
    if (c < CDBL) out[c] = acc[i];
  }
}

// -------------------- Kernel 4: selective scan ----------------------------
// One wave32 per (b,k, 32 channels of Din). Each lane owns h[16] state.
// Fuses: low-rank dt expansion, softplus(+bias), recurrence, y, D*u skip.
// All cross-lane broadcasts use v_readlane (constant source lanes).
__global__ __launch_bounds__(32) void k_scan(const float* __restrict__ xc,
                                             const float* __restrict__ xdbl,
                                             const float* __restrict__ wdt,
                                             const float* __restrict__ dtb,
                                             const float* __restrict__ alog,
                                             const float* __restrict__ Ds,
                                             float* __restrict__ ys) {
  const int lane = threadIdx.x;
  const int dg = blockIdx.x % (DIN/32);
  const int k  = (blockIdx.x / (DIN/32)) % KDIR;
  const int b  =  blockIdx.x / ((DIN/32) * KDIR);
  const int d  = dg*32 + lane;
  const int kd = k*DIN + d;

  float Arow[NSTATE], hst[NSTATE], w6[RR];
#pragma unroll
  for (int n = 0; n < NSTATE; ++n) {
    Arow[n] = -__expf(alog[(size_t)kd*NSTATE + n]);
    hst[n]  = 0.f;
  }
#pragma unroll
  for (int r = 0; r < RR; ++r) w6[r] = wdt[(size_t)kd*RR + r];
  const float bias = dtb[kd];
  const float dsk  = Ds[kd];

  const float* xdb = xdbl + (size_t)(b*KDIR + k) * LL * CDBL;
  const float* xcb = xc   + (size_t)b * LL * DIN;
  float*       ysb = ys   + (size_t)(b*KDIR + k) * LL * DIN;

  for (int l = 0; l < LL; ++l) {
    const int pos = dir_pos(k, l);
    const float du = xcb[(size_t)pos*DIN + d];
    const float v1 = xdb[(size_t)l*CDBL + lane];
    const float v2 = (lane < RR) ? xdb[(size_t)l*CDBL + 32 + lane] : 0.f;
    if (l + 16 < LL)   // keep the 38-ch stream warm in cache
      __builtin_prefetch(xdb + (size_t)(l + 16)*CDBL + lane, 0, 1);

    float dtv = bias;
#pragma unroll
    for (int r = 0; r < RR; ++r) dtv = fmaf(w6[r], bcast(v1, r), dtv);
    dtv = (dtv > 20.f) ? dtv : log1pf(__expf(dtv));   // softplus
    const float dtu = dtv * du;

    float y = 0.f;
#pragma unroll
    for (int n = 0; n < NSTATE; ++n) {
      const float Bn = bcast(v1, RR + n);
      const float Cn = (n < 10) ? bcast(v1, 22 + n)
                                : bcast(v2, n - 10);
      const float dA = __expf(dtv * Arow[n]);
      hst[n] = fmaf(dA, hst[n], dtu * Bn);
      y = fmaf(hst[n], Cn, y);
    }
    ysb[(size_t)l*DIN + d] = fmaf(dsk, du, y);
  }
}

// -------------------- Kernel 5: cross-merge + LayerNorm + gate ------------
__global__ __launch_bounds__(DIN) void k_merge(const float* __restrict__ ys,
                                               const float* __restrict__ xz,
                                               const float* __restrict__ gamma,
                                               const float* __restrict__ beta,
                                               float* __restrict__ g) {
  __shared__ float red[2][8];
  const int bl = blockIdx.x;                  // b*L + l
  const int b = bl / LL, l = bl % LL;
  const int hh = l / WW, ww = l % WW;
  const int d = threadIdx.x;
  const int s1 = ww*HH + hh;                  // transposed-scan sequence index
  const size_t base = (size_t)b * KDIR * LL * DIN;
  float v = ys[base + ((size_t)0*LL + l)            * DIN + d]
          + ys[base + ((size_t)2*LL + (LL-1-l))     * DIN + d]
          + ys[base + ((size_t)1*LL + s1)           * DIN + d]
          + ys[base + ((size_t)3*LL + (LL-1-s1))    * DIN + d];

  float s = v, s2 = v*v;
#pragma unroll
  for (int off = 16; off > 0; off >>= 1) {
    s  += __shfl_down(s,  off, 32);
    s2 += __shfl_down(s2, off, 32);
  }
  const int wid = threadIdx.x >> 5, lane = threadIdx.x & 31;
  if (lane == 0) { red[0][wid] = s; red[1][wid] = s2; }
  __syncthreads();
  float ts = 0.f, ts2 = 0.f;
#pragma unroll
  for (int i = 0; i < 6; ++i) { ts += red[0][i]; ts2 += red[1][i]; }
  const float mu  = ts / DIN;
  const float var = ts2 / DIN - mu*mu;
  const float yhat = (v - mu) * rsqrtf(var + 1e-5f) * gamma[d] + beta[d];
  const float z = xz[(size_t)bl * (2*DIN) + DIN + d];
  g[(size_t)bl*DIN + d] = yhat * (z / (1.f + __expf(-z)));
}

// -------------------- Kernel 6: out_proj GEMM -----------------------------
// out[M,96] = g[M,192] @ out_proj_w[96,192]^T
__global__ __launch_bounds__(256) void k_outproj(const float* __restrict__ g,
                                                 const float* __restrict__ w,
                                                 float* __restrict__ out) {
  const int lane = threadIdx.x & 31;
  const int wid  = blockIdx.x * 8 + (threadIdx.x >> 5);
  const int TN = CC/16;                        // 6 col tiles
  const int tn = wid % TN, tm = wid / TN;
  const int row0 = tm*16, col0 = tn*16;
  const int mrow = lane & 15, half = lane >> 4;
  const float* arow = g + (size_t)(row0 + mrow) * DIN;
  const float* brow = w + (size_t)(col0 + mrow) * DIN;
  v8f acc = {};
#pragma unroll 8
  for (int k0 = 0; k0 < DIN; k0 += 4) {
    v2f a = *(const v2f*)(arow + k0 + 2*half);
    v2f b = *(const v2f*)(brow + k0 + 2*half);
    acc = wmma_f32_k4(a, b, acc);
  }
  float* o = out + (size_t)row0 * CC + col0 + mrow;
#pragma unroll
  for (int i = 0; i < 8; ++i)
    o[(size_t)(8*half + i) * CC] = acc[i];
}

// ---------------------------------------------------------------------------
extern "C" void kernel_launch(void* const* d_in, const int* in_sizes, int n_in,
                              void* d_out, int out_size, void* d_ws, size_t ws_size,
                              hipStream_t stream) {
  const float* x    = (const float*)d_in[0];   // [B,H,W,C]
  const float* inw  = (const float*)d_in[1];   // [384,96]
  const float* cw   = (const float*)d_in[2];   // [192,1,3,3]
  const float* cb   = (const float*)d_in[3];   // [192]
  const float* xpw  = (const float*)d_in[4];   // [4,38,192]
  const float* dtw  = (const float*)d_in[5];   // [4,192,6]
  const float* dtbi = (const float*)d_in[6];   // [4,192]
  const float* alog = (const float*)d_in[7];   // [768,16]
  const float* Ds   = (const float*)d_in[8];   // [768]
  const float* gam  = (const float*)d_in[9];   // [192]
  const float* bet  = (const float*)d_in[10];  // [192]
  const float* ow   = (const float*)d_in[11];  // [96,192]
  float* out = (float*)d_out;

  // workspace layout (fp32), total ~221 MB
  float* ws = (float*)d_ws;
  size_t o = 0;
  float* xz   = ws + o; o += (size_t)MROWS * (2*DIN);          // 50.3 MB
  float* xc   = ws + o; o += (size_t)MROWS * DIN;              // 25.2 MB
  float* xdbl = ws + o; o += (size_t)BB * KDIR * LL * CDBL;    // 19.9 MB
  float* ysb  = ws + o; o += (size_t)BB * KDIR * LL * DIN;     // 100.7 MB
  float* g    = ws + o; o += (size_t)MROWS * DIN;              // 25.2 MB
  (void)ws_size; (void)in_sizes; (void)n_in; (void)out_size;

  k_inproj <<<(MROWS/16) * ((2*DIN)/16) / 8, 256, 0, stream>>>(x, inw, xz);
  k_conv   <<<(MROWS*DIN) / 256,             256, 0, stream>>>(xz, cw, cb, xc);
  k_xdbl   <<<(BB*KDIR*3*(LL/16)) / 8,       256, 0, stream>>>(xc, xpw, xdbl);
  k_scan   <<<BB*KDIR*(DIN/32),               32, 0, stream>>>(xc, xdbl, dtw, dtbi,
                                                               alog, Ds, ysb);
  k_merge  <<<MROWS,                         DIN, 0, stream>>>(ysb, xz, gam, bet, g);
  k_outproj<<<(MROWS/16) * (CC/16) / 8,      256, 0, stream>>>(g, ow, out);
}